// RWKV_Tmix_x060c_37864431682266
// MI455X (gfx1250) — compile-verified
//
#include <hip/hip_runtime.h>
#include <hip/hip_bf16.h>
#include <math.h>

// ---------------- problem constants ----------------
#define BB   32
#define CC   2048
#define DD   2048
#define HH   32
#define SS   64
#define ROWS 1584          // (2+S)*L
#define I1R  793           // token-shift row
#define R0R  794           // wkv state slice start
#define DMIX 32
#define DDEC 64
#define LN_EPS 1e-5f

typedef float v2f __attribute__((ext_vector_type(2)));
typedef float v4f __attribute__((ext_vector_type(4)));
typedef float v8f __attribute__((ext_vector_type(8)));

// ---------------- 0) full state copy (new_state = state) ----------------
// Pure streaming traffic (830 MB round trip vs 192 MB L2): use non-temporal
// load/store policy so the copy does not evict the GEMM weights from L2.
__global__ void copy_state_kernel(const v4f* __restrict__ src,
                                  v4f* __restrict__ dst, long n4) {
    long i = (long)blockIdx.x * blockDim.x + threadIdx.x;
    long stride = (long)gridDim.x * blockDim.x;
    for (; i < n4; i += stride) {
        v4f t = __builtin_nontemporal_load(src + i);
        __builtin_nontemporal_store(t, dst + i);
    }
}

// ---------------- 1) token-shift + LoRA mix weights ----------------
// one block per batch row b; block = 256 threads
__global__ void mix_kernel(const float* __restrict__ x,
                           const float* __restrict__ state,
                           const float* __restrict__ tmx,
                           const float* __restrict__ tmw,
                           const float* __restrict__ tmk,
                           const float* __restrict__ tmv,
                           const float* __restrict__ tmr,
                           const float* __restrict__ maa_w1,   // [C, 128]
                           const float* __restrict__ maa_w2,   // [4, 32, C]
                           float* __restrict__ out_ns,         // new_state base
                           float* __restrict__ xr,
                           float* __restrict__ xk,
                           float* __restrict__ xv,
                           float* __restrict__ xw) {
    __shared__ float s_sx[CC];
    __shared__ float s_xxx[CC];
    __shared__ float s_part[2][4 * DMIX];   // split dot partials
    __shared__ float s_mix[4 * DMIX];

    const int b   = blockIdx.x;
    const int tid = threadIdx.x;
    const float* xb  = x + (long)b * CC;
    const float* stb = state + ((long)b * ROWS + I1R) * CC;
    float* nsb = out_ns + ((long)b * ROWS + I1R) * CC;

    for (int c = tid; c < CC; c += blockDim.x) {
        float xv_ = xb[c];
        float sx  = stb[c] - xv_;
        s_sx[c]   = sx;
        s_xxx[c]  = xv_ + sx * tmx[c];
        nsb[c]    = xv_;                       // new_state token-shift row
    }
    __syncthreads();

    // mix = tanh(xxx @ maa_w1): 128 dots of length 2048, split into 2 chunks
    {
        const int j     = tid & 127;           // output index
        const int chunk = tid >> 7;            // 0 or 1
        const int c0    = chunk * (CC / 2);
        float acc = 0.f;
        for (int c = c0; c < c0 + CC / 2; ++c)
            acc += s_xxx[c] * maa_w1[(long)c * (4 * DMIX) + j];
        s_part[chunk][j] = acc;
    }
    __syncthreads();
    if (tid < 4 * DMIX)
        s_mix[tid] = tanhf(s_part[0][tid] + s_part[1][tid]);
    __syncthreads();

    for (int c = tid; c < CC; c += blockDim.x) {
        float m[4];
        #pragma unroll
        for (int f = 0; f < 4; ++f) {
            float acc = 0.f;
            #pragma unroll 8
            for (int d = 0; d < DMIX; ++d)
                acc += s_mix[f * DMIX + d] * maa_w2[((long)(f * DMIX + d)) * CC + c];
            m[f] = acc;
        }
        float xv_ = xb[c];
        float sx  = s_sx[c];
        long  o   = (long)b * CC + c;
        xr[o] = xv_ + sx * (tmr[c] + m[0]);
        xk[o] = xv_ + sx * (tmk[c] + m[1]);
        xv[o] = xv_ + sx * (tmv[c] + m[2]);
        xw[o] = xv_ + sx * (tmw[c] + m[3]);
    }
}

// ---------------- 2) f32 WMMA GEMM: Y[32,N] = X[32,K] @ W[K,N] ----------------
// block = 256 threads (8 waves); each wave owns 4 adjacent 16x16 tiles along N.
__device__ __forceinline__ v8f wmma_f32(v2f a, v2f b, v8f c) {
    return __builtin_amdgcn_wmma_f32_16x16x4_f32(
        /*neg_a=*/false, a, /*neg_b=*/false, b,
        /*c_mod=*/(short)0, c, /*reuse_a=*/false, /*reuse_b=*/false);
}

__global__ void gemm_wmma_kernel(const float* __restrict__ X,
                                 const float* __restrict__ W,
                                 float* __restrict__ Y,
                                 int K, int N) {
    const int lane = threadIdx.x & 31;
    const int wave = threadIdx.x >> 5;                 // 0..7
    const int row  = lane & 15;                        // M (for A) / N (for B) index
    const int kh   = (lane >> 4) << 1;                 // K half-select: 0 or 2
    const int m0   = blockIdx.y * 16;
    const int n0   = (blockIdx.x * 8 + wave) * 64;     // 4 tiles of 16

    v8f acc[4];
    #pragma unroll
    for (int t = 0; t < 4; ++t) acc[t] = (v8f){0.f,0.f,0.f,0.f,0.f,0.f,0.f,0.f};

    const float* xrow = X + (long)(m0 + row) * K;

    for (int k0 = 0; k0 < K; k0 += 4) {
        v2f a;
        a.x = xrow[k0 + kh];
        a.y = xrow[k0 + kh + 1];
        const float* w0 = W + (long)(k0 + kh) * N + n0 + row;
        const float* w1 = W + (long)(k0 + kh + 1) * N + n0 + row;
        #pragma unroll
        for (int t = 0; t < 4; ++t) {
            v2f bfrag;
            bfrag.x = w0[t * 16];
            bfrag.y = w1[t * 16];
            acc[t] = wmma_f32(a, bfrag, acc[t]);
        }
    }

    const int mbase = m0 + ((lane >> 4) << 3);         // +8 for upper lanes
    const int nlane = lane & 15;
    #pragma unroll
    for (int t = 0; t < 4; ++t) {
        #pragma unroll
        for (int i = 0; i < 8; ++i) {
            Y[(long)(mbase + i) * N + n0 + t * 16 + nlane] = acc[t][i];
        }
    }
}

// ---------------- 3) decay LoRA: w = exp(-exp(td + tanh(xw@w1)@w2)) ---------
__global__ void decay_kernel(const float* __restrict__ xw,     // [B,C]
                             const float* __restrict__ tdec,   // [D]
                             const float* __restrict__ td_w1,  // [C,64]
                             const float* __restrict__ td_w2,  // [64,D]
                             float* __restrict__ k,            // [B,D] in/out
                             float* __restrict__ w) {          // [B,D] out
    __shared__ float s_part[4][DDEC];
    __shared__ float s_t[DDEC];
    const int b   = blockIdx.x;
    const int tid = threadIdx.x;
    const float* xwb = xw + (long)b * CC;

    // 64 dots of length 2048, split into 4 chunks of 512
    {
        const int j     = tid & 63;
        const int chunk = tid >> 6;            // 0..3
        const int c0    = chunk * (CC / 4);
        float acc = 0.f;
        for (int c = c0; c < c0 + CC / 4; ++c)
            acc += xwb[c] * td_w1[(long)c * DDEC + j];
        s_part[chunk][j] = acc;
    }
    __syncthreads();
    if (tid < DDEC)
        s_t[tid] = tanhf(s_part[0][tid] + s_part[1][tid] +
                         s_part[2][tid] + s_part[3][tid]);
    __syncthreads();

    for (int c = tid; c < DD; c += blockDim.x) {
        float z = tdec[c];
        #pragma unroll 16
        for (int j = 0; j < DDEC; ++j)
            z += s_t[j] * td_w2[(long)j * DD + c];
        float wv = expf(-expf(z));
        long o = (long)b * DD + c;
        w[o] = wv;
        k[o] *= (1.0f - expf(-expf(wv)));              // faithful to source
    }
}

// ---------------- 4) per-head linear-attention state update ----------------
// grid = (H, B), block = 64 (thread j = output column within head)
__global__ void attn_kernel(const float* __restrict__ r,       // [B,D]
                            const float* __restrict__ k,
                            const float* __restrict__ v,
                            const float* __restrict__ w,
                            const float* __restrict__ fa,      // [H,S]
                            const float* __restrict__ state,   // input state
                            float* __restrict__ out_ns,        // new_state base
                            float* __restrict__ o) {           // [B,D]
    __shared__ float s_r[SS], s_k[SS], s_w[SS], s_f[SS];
    const int h = blockIdx.x;
    const int b = blockIdx.y;
    const int j = threadIdx.x;                         // 0..63

    const long hb = (long)b * DD + h * SS;
    s_r[j] = r[hb + j];
    s_k[j] = k[hb + j];
    s_w[j] = w[hb + j];
    s_f[j] = fa[h * SS + j];
    __syncthreads();

    const float vj = v[hb + j];
    const long sbase = ((long)b * ROWS + R0R + 2 * h) * CC;

    float oj = 0.f;
    #pragma unroll 4
    for (int i = 0; i < SS; ++i) {
        long off = (long)(i >> 5) * CC + (i & 31) * SS + j;
        float s  = state[sbase + off];
        float a  = s_k[i] * vj;                        // outer product elem
        oj += s_r[i] * (s_f[i] * a + s);
        out_ns[sbase + off] = a + s_w[i] * s;
    }
    o[hb + j] = oj;
}

// ---------------- 5) LayerNorm over D ----------------
__global__ void ln_kernel(const float* __restrict__ o,
                          const float* __restrict__ g,
                          const float* __restrict__ bta,
                          float* __restrict__ xn) {
    __shared__ float s_sum[256];
    __shared__ float s_sq[256];
    const int b   = blockIdx.x;
    const int tid = threadIdx.x;
    const float* ob = o + (long)b * DD;

    float sum = 0.f, sq = 0.f;
    for (int c = tid; c < DD; c += blockDim.x) {
        float t = ob[c];
        sum += t; sq += t * t;
    }
    s_sum[tid] = sum; s_sq[tid] = sq;
    __syncthreads();
    for (int s = 128; s > 0; s >>= 1) {
        if (tid < s) { s_sum[tid] += s_sum[tid + s]; s_sq[tid] += s_sq[tid + s]; }
        __syncthreads();
    }
    const float mu   = s_sum[0] / DD;
    const float var  = s_sq[0] / DD - mu * mu;
    const float rstd = rsqrtf(var + LN_EPS);

    for (int c = tid; c < DD; c += blockDim.x)
        xn[(long)b * DD + c] = (ob[c] - mu) * rstd * g[c] + bta[c];
}

// ---------------- launcher ----------------
extern "C" void kernel_launch(void* const* d_in, const int* in_sizes, int n_in,
                              void* d_out, int out_size, void* d_ws, size_t ws_size,
                              hipStream_t stream) {
    const float* x       = (const float*)d_in[0];
    const float* state   = (const float*)d_in[1];
    const float* tmx     = (const float*)d_in[2];
    const float* tmw     = (const float*)d_in[3];
    const float* tmk     = (const float*)d_in[4];
    const float* tmv     = (const float*)d_in[5];
    const float* tmr     = (const float*)d_in[6];
    const float* maa_w1  = (const float*)d_in[7];
    const float* maa_w2  = (const float*)d_in[8];
    const float* tdec    = (const float*)d_in[9];
    const float* td_w1   = (const float*)d_in[10];
    const float* td_w2   = (const float*)d_in[11];
    const float* faaaa   = (const float*)d_in[12];
    const float* Wr      = (const float*)d_in[13];
    const float* Wk      = (const float*)d_in[14];
    const float* Wv      = (const float*)d_in[15];
    const float* Wo      = (const float*)d_in[16];
    const float* ln_g    = (const float*)d_in[17];
    const float* ln_b    = (const float*)d_in[18];

    float* out    = (float*)d_out;                 // [B,C] output
    float* out_ns = out + (long)BB * CC;           // new_state [B,ROWS,C]

    const long BC = (long)BB * CC;
    float* ws  = (float*)d_ws;
    float* xr  = ws + 0 * BC;
    float* xk  = ws + 1 * BC;
    float* xvv = ws + 2 * BC;
    float* xw  = ws + 3 * BC;
    float* rr  = ws + 4 * BC;
    float* kk  = ws + 5 * BC;
    float* vv  = ws + 6 * BC;
    float* ww  = ws + 7 * BC;
    float* oo  = ws + 8 * BC;
    float* xn  = ws + 9 * BC;

    // 0) copy state -> new_state (bulk of the traffic; NT-hinted B128 streams)
    long n4 = ((long)BB * ROWS * CC) / 4;
    copy_state_kernel<<<8192, 256, 0, stream>>>((const v4f*)state,
                                                (v4f*)out_ns, n4);

    // 1) token-shift + LoRA mix, produces xr/xk/xv/xw and new_state row I1
    mix_kernel<<<BB, 256, 0, stream>>>(x, state, tmx, tmw, tmk, tmv, tmr,
                                       maa_w1, maa_w2, out_ns, xr, xk, xvv, xw);

    // 2) r/k/v projections (WMMA f32 16x16x4)
    dim3 ggrid(DD / 512, BB / 16);
    gemm_wmma_kernel<<<ggrid, 256, 0, stream>>>(xr,  Wr, rr, CC, DD);
    gemm_wmma_kernel<<<ggrid, 256, 0, stream>>>(xk,  Wk, kk, CC, DD);
    gemm_wmma_kernel<<<ggrid, 256, 0, stream>>>(xvv, Wv, vv, CC, DD);

    // 3) decay LoRA + k scaling
    decay_kernel<<<BB, 256, 0, stream>>>(xw, tdec, td_w1, td_w2, kk, ww);

    // 4) per-head attention state update (writes R0:R1 slice of new_state)
    attn_kernel<<<dim3(HH, BB), SS, 0, stream>>>(rr, kk, vv, ww, faaaa,
                                                 state, out_ns, oo);

    // 5) LayerNorm over D
    ln_kernel<<<BB, 256, 0, stream>>>(oo, ln_g, ln_b, xn);

    // 6) output projection (WMMA) -> d_out[0 : B*C]
    gemm_wmma_kernel<<<dim3(CC / 512, BB / 16), 256, 0, stream>>>(xn, Wo, out, DD, CC);
}